// InteractionBlock_31671088840961
// MI455X (gfx1250) — compile-verified
//
#include <hip/hip_runtime.h>
#include <hip/hip_bf16.h>
#include <math.h>

// PaiNN-style interaction block for MI455X (gfx1250, wave32).
// Memory-bound (AI ~4 FLOP/byte vs 23.3 TB/s HBM) -> stay f32, use
// V_WMMA_F32_16X16X4_F32 for both GEMMs. W_filter is DMA'd to LDS with the
// Tensor Data Mover (TENSOR_LOAD_TO_LDS) and its fragments are register-
// resident across 4 edge M-tiles per block.

typedef __attribute__((ext_vector_type(2))) float v2f;
typedef __attribute__((ext_vector_type(8))) float v8f;
typedef __attribute__((ext_vector_type(4))) unsigned int v4u;
typedef __attribute__((ext_vector_type(4))) int v4i;
typedef __attribute__((ext_vector_type(8))) int v8i;

#define CUTOFF 5.0f
#define DD 64          // feature dim
#define TD 192         // 3*D
#define RR 20          // radial basis size
#define WAVES 12
#define BLK (WAVES * 32)
#define EPB 64         // edges per block (4 WMMA M-tiles)
#define SQRT_2_OVER_C 0.63245553203367586f  // sqrt(2/5)

__device__ __forceinline__ float siluf(float x) { return x / (1.0f + __expf(-x)); }

__device__ __forceinline__ float envf(float d) {
  float u = d * (1.0f / CUTOFF);
  float u2 = u * u, u3 = u2 * u;
  float u6 = u3 * u3, u7 = u6 * u, u8 = u7 * u;
  float e = 1.0f - 28.0f * u6 + 48.0f * u7 - 21.0f * u8;
  return d < CUTOFF ? e : 0.0f;
}

// TDM: async-DMA a 2D f32 tile [dim1 rows x dim0 cols, row stride stride0]
// from global memory into LDS. D# layout per CDNA5 ISA ch.8 (group0/group1).
__device__ __forceinline__ void tdm_load_2d_f32(const float* gptr, unsigned ldsAddr,
                                                unsigned dim0, unsigned dim1,
                                                unsigned stride0) {
  unsigned long long ga = (unsigned long long)(uintptr_t)gptr;
  v4u g0;
  g0[0] = 1u;                                              // count=1 (valid), user mode
  g0[1] = ldsAddr;                                         // lds_addr [63:32]
  g0[2] = (unsigned)(ga & 0xFFFFFFFFu);                    // global_addr [95:64]
  g0[3] = (unsigned)((ga >> 32) & 0x01FFFFFFu) | (2u << 30); // ga[56:32] | type=2
  v8i g1;
  g1[0] = (int)(2u << 16);                                 // data_size=2 (4B), mask=0
  g1[1] = (int)((dim0 & 0xFFFFu) << 16);                   // tensor_dim0 lo16 @bit48
  g1[2] = (int)(((dim0 >> 16) & 0xFFFFu) | ((dim1 & 0xFFFFu) << 16)); // dim0 hi | dim1 lo
  g1[3] = (int)(((dim1 >> 16) & 0xFFFFu) | ((dim0 & 0xFFFFu) << 16)); // dim1 hi | tile_dim0
  g1[4] = (int)(dim1 & 0xFFFFu);                           // tile_dim1 | tile_dim2=0
  g1[5] = (int)stride0;                                    // tensor_dim0_stride lo32
  g1[6] = 0;                                               // stride0 hi | stride1 lo
  g1[7] = 0;                                               // stride1 hi
  v4i gz = {0, 0, 0, 0};
#if defined(__clang_major__) && (__clang_major__ >= 23)
  v8i gz8 = {0, 0, 0, 0, 0, 0, 0, 0};
  __builtin_amdgcn_tensor_load_to_lds(g0, g1, gz, gz, gz8, 0);
#else
  __builtin_amdgcn_tensor_load_to_lds(g0, g1, gz, gz, 0);
#endif
}

// ---------------------------------------------------------------- zero output
__global__ void k_zero(float* __restrict__ p, long long n) {
  long long i = (long long)blockIdx.x * blockDim.x + threadIdx.x;
  long long stride = (long long)gridDim.x * blockDim.x;
  for (; i < n; i += stride) p[i] = 0.0f;
}

// ------------------------------------------------- per-atom MLP: h[N,192]
// h = silu(se @ W1 + b1) @ W2 + b2 ; 16 atoms per block, f32 WMMA 16x16x4.
__global__ __launch_bounds__(BLK) void k_phi(
    const float* __restrict__ se, const float* __restrict__ W1,
    const float* __restrict__ b1, const float* __restrict__ W2,
    const float* __restrict__ b2, float* __restrict__ h) {
  __shared__ float sA[16][DD + 1];  // +1 pad: avoid 64-bank conflicts
  __shared__ float sT[16][DD + 1];
  const int aBase = blockIdx.x * 16;

  for (int i = threadIdx.x; i < 16 * DD; i += BLK)
    sA[i / DD][i % DD] = se[(size_t)(aBase + i / DD) * DD + (i % DD)];
  __syncthreads();

  const int w = threadIdx.x >> 5, lane = threadIdx.x & 31;
  const int half = lane >> 4, n = lane & 15, mrow = lane & 15;

  // stage 1: t = silu(se @ W1 + b1), 64 cols -> waves 0..3 (one 16-col tile each)
  if (w < 4) {
    v8f acc = {};
#pragma unroll
    for (int kk = 0; kk < 16; ++kk) {
      const int k0 = 4 * kk + 2 * half;
      v2f a; a.x = sA[mrow][k0]; a.y = sA[mrow][k0 + 1];
      v2f b; b.x = W1[k0 * DD + w * 16 + n]; b.y = W1[(k0 + 1) * DD + w * 16 + n];
      acc = __builtin_amdgcn_wmma_f32_16x16x4_f32(false, a, false, b, (short)0,
                                                  acc, false, false);
    }
#pragma unroll
    for (int r = 0; r < 8; ++r) {
      const int m = half * 8 + r, col = w * 16 + n;
      sT[m][col] = siluf(acc[r] + b1[col]);
    }
  }
  __syncthreads();

  // stage 2: h = t @ W2 + b2, 192 cols -> all 12 waves
  {
    v8f acc = {};
#pragma unroll
    for (int kk = 0; kk < 16; ++kk) {
      const int k0 = 4 * kk + 2 * half;
      v2f a; a.x = sT[mrow][k0]; a.y = sT[mrow][k0 + 1];
      v2f b; b.x = W2[k0 * TD + w * 16 + n]; b.y = W2[(k0 + 1) * TD + w * 16 + n];
      acc = __builtin_amdgcn_wmma_f32_16x16x4_f32(false, a, false, b, (short)0,
                                                  acc, false, false);
    }
#pragma unroll
    for (int r = 0; r < 8; ++r) {
      const int m = half * 8 + r, col = w * 16 + n;
      h[(size_t)(aBase + m) * TD + col] = acc[r] + b2[col];
    }
  }
}

// ----------------------------------------- per-edge: filter GEMM + scatter
// 64 edges per block; W_filter DMA'd to LDS via TDM; each of 12 waves keeps
// its 5 B-fragments in registers across the 4 edge M-tiles.
__global__ __launch_bounds__(BLK) void k_edge(
    const float* __restrict__ nv, const int* __restrict__ central,
    const int* __restrict__ neigh, const float* __restrict__ freqs,
    const float* __restrict__ Wf, const float* __restrict__ bfil,
    const float* __restrict__ h, const float* __restrict__ ve,
    float* __restrict__ dv, float* __restrict__ ds) {
  __shared__ float sWf[RR * TD];     // 15360 B, TDM destination
  __shared__ float sBes[EPB][RR];
  __shared__ float sUnit[EPB][3];
  __shared__ float sEnv[EPB];
  __shared__ float sDist[EPB];
  __shared__ int sCent[EPB];
  __shared__ int sNeigh[EPB];
  __shared__ float sX[EPB][TD];      // 49152 B

  const int eBase = blockIdx.x * EPB;
  const int tid = threadIdx.x;
  const int w = tid >> 5, lane = tid & 31;

  // kick off the W_filter DMA first so it overlaps the per-edge VALU setup
  if (w == 0) {
    tdm_load_2d_f32(Wf, (unsigned)(uintptr_t)(void*)sWf, TD, RR, TD);
  }

  if (tid < EPB) {
    const int e = eBase + tid;
    const float x = nv[3 * (size_t)e + 0];
    const float y = nv[3 * (size_t)e + 1];
    const float z = nv[3 * (size_t)e + 2];
    const float d = sqrtf(x * x + y * y + z * z);
    const float inv = 1.0f / d;
    sDist[tid] = d;
    sUnit[tid][0] = x * inv; sUnit[tid][1] = y * inv; sUnit[tid][2] = z * inv;
    sEnv[tid] = envf(d);
    sCent[tid] = central[e];
    sNeigh[tid] = neigh[e];
  }
  __syncthreads();
  for (int i = tid; i < EPB * RR; i += BLK) {
    const int m = i / RR, r = i % RR;
    const float d = sDist[m];
    sBes[m][r] = SQRT_2_OVER_C * sinf(freqs[r] * d) / d;
  }
  if (w == 0) __builtin_amdgcn_s_wait_tensorcnt(0);  // W_filter resident in LDS
  __syncthreads();

  const int half = lane >> 4, n = lane & 15, mrow = lane & 15;

  // B fragments (W_filter col tile w) loaded once, reused for all 4 M-tiles
  v2f bfr[5];
#pragma unroll
  for (int kk = 0; kk < 5; ++kk) {
    const int k0 = 4 * kk + 2 * half;
    bfr[kk].x = sWf[k0 * TD + w * 16 + n];
    bfr[kk].y = sWf[(k0 + 1) * TD + w * 16 + n];
  }
  const float bias = bfil[w * 16 + n];

#pragma unroll
  for (int mt = 0; mt < EPB / 16; ++mt) {
    v8f acc = {};
#pragma unroll
    for (int kk = 0; kk < 5; ++kk) {
      const int k0 = 4 * kk + 2 * half;
      v2f a; a.x = sBes[mt * 16 + mrow][k0]; a.y = sBes[mt * 16 + mrow][k0 + 1];
      acc = __builtin_amdgcn_wmma_f32_16x16x4_f32(false, a, false, bfr[kk],
                                                  (short)0, acc, false, false);
    }
#pragma unroll
    for (int r = 0; r < 8; ++r) {
      const int m = mt * 16 + half * 8 + r, col = w * 16 + n;
      const float f = (acc[r] + bias) * sEnv[m];
      sX[m][col] = f * h[(size_t)sCent[m] * TD + col];  // x_ij = filt * h[central]
    }
  }
  __syncthreads();

  // scatter: delta_s += a ; delta_v += b*unit + c*ve[neigh]  (fused, 4 atomics)
  for (int i = tid; i < EPB * DD; i += BLK) {
    const int m = i >> 6, j = i & 63;
    const int ng = sNeigh[m];
    const float av = sX[m][j];
    const float bv = sX[m][DD + j];
    const float cv = sX[m][2 * DD + j];
    atomicAdd(&ds[(size_t)ng * DD + j], av);
    const size_t base = ((size_t)ng * DD + j) * 3;
    const float* vp = ve + base;
    atomicAdd(&dv[base + 0], bv * sUnit[m][0] + cv * vp[0]);
    atomicAdd(&dv[base + 1], bv * sUnit[m][1] + cv * vp[1]);
    atomicAdd(&dv[base + 2], bv * sUnit[m][2] + cv * vp[2]);
  }
}

// ------------------------------------------------- scalar tails (rarely used)
__global__ void k_phi_tail(const float* __restrict__ se, const float* __restrict__ W1,
                           const float* __restrict__ b1, const float* __restrict__ W2,
                           const float* __restrict__ b2, float* __restrict__ h,
                           int aBase) {
  const int a = aBase + blockIdx.x;
  const int c = threadIdx.x;  // 0..191
  float acc = b2[c];
  for (int j = 0; j < DD; ++j) {
    float t = b1[j];
    for (int k = 0; k < DD; ++k) t += se[(size_t)a * DD + k] * W1[k * DD + j];
    acc += siluf(t) * W2[j * TD + c];
  }
  h[(size_t)a * TD + c] = acc;
}

__global__ void k_edge_tail(const float* __restrict__ nv, const int* __restrict__ central,
                            const int* __restrict__ neigh, const float* __restrict__ freqs,
                            const float* __restrict__ Wf, const float* __restrict__ bfil,
                            const float* __restrict__ h, const float* __restrict__ ve,
                            float* __restrict__ dv, float* __restrict__ ds, int eBase) {
  __shared__ float bes[RR];
  __shared__ float unit[3];
  __shared__ float envv, dd;
  __shared__ int cg, ng;
  const int e = eBase + blockIdx.x;
  if (threadIdx.x == 0) {
    const float x = nv[3 * (size_t)e], y = nv[3 * (size_t)e + 1], z = nv[3 * (size_t)e + 2];
    const float d = sqrtf(x * x + y * y + z * z);
    dd = d; const float inv = 1.0f / d;
    unit[0] = x * inv; unit[1] = y * inv; unit[2] = z * inv;
    envv = envf(d); cg = central[e]; ng = neigh[e];
  }
  __syncthreads();
  if (threadIdx.x < RR)
    bes[threadIdx.x] = SQRT_2_OVER_C * sinf(freqs[threadIdx.x] * dd) / dd;
  __syncthreads();
  const int c = threadIdx.x;  // 0..191
  float f = bfil[c];
  for (int r = 0; r < RR; ++r) f += bes[r] * Wf[r * TD + c];
  f *= envv;
  const float x = f * h[(size_t)cg * TD + c];
  if (c < DD) {
    atomicAdd(&ds[(size_t)ng * DD + c], x);
  } else if (c < 2 * DD) {
    const int j = c - DD;
    const size_t base = ((size_t)ng * DD + j) * 3;
    for (int k = 0; k < 3; ++k) atomicAdd(&dv[base + k], x * unit[k]);
  } else {
    const int j = c - 2 * DD;
    const size_t base = ((size_t)ng * DD + j) * 3;
    for (int k = 0; k < 3; ++k) atomicAdd(&dv[base + k], x * ve[base + k]);
  }
}

// ---------------------------------------------------------------------- host
extern "C" void kernel_launch(void* const* d_in, const int* in_sizes, int n_in,
                              void* d_out, int out_size, void* d_ws, size_t ws_size,
                              hipStream_t stream) {
  const float* ve      = (const float*)d_in[0];
  const float* se      = (const float*)d_in[1];
  const float* nv      = (const float*)d_in[2];
  const int*   central = (const int*)d_in[3];
  const int*   neigh   = (const int*)d_in[4];
  const float* freqs   = (const float*)d_in[5];
  const float* Wf      = (const float*)d_in[6];
  const float* bfil    = (const float*)d_in[7];
  const float* W1      = (const float*)d_in[8];
  const float* b1      = (const float*)d_in[9];
  const float* W2      = (const float*)d_in[10];
  const float* b2      = (const float*)d_in[11];

  const int NA = in_sizes[1] / DD;  // atoms
  const int E  = in_sizes[3];       // edges

  float* out = (float*)d_out;
  float* dv = out;                           // delta_v [NA, 64, 3]
  float* ds = out + (size_t)NA * DD * 3;     // delta_s [NA, 64]
  float* h  = (float*)d_ws;                  // [NA, 192] intermediate

  k_zero<<<2048, 256, 0, stream>>>(out, (long long)out_size);

  const int aB = NA / 16, aR = NA % 16;
  if (aB) k_phi<<<aB, BLK, 0, stream>>>(se, W1, b1, W2, b2, h);
  if (aR) k_phi_tail<<<aR, TD, 0, stream>>>(se, W1, b1, W2, b2, h, NA - aR);

  const int eB = E / EPB, eR = E % EPB;
  if (eB) k_edge<<<eB, BLK, 0, stream>>>(nv, central, neigh, freqs, Wf, bfil, h, ve, dv, ds);
  if (eR) k_edge_tail<<<eR, TD, 0, stream>>>(nv, central, neigh, freqs, Wf, bfil, h, ve,
                                             dv, ds, E - eR);
}